// GCNConv_61022895341671
// MI455X (gfx1250) — compile-verified
//
#include <hip/hip_runtime.h>
#include <math.h>

// RGCN layer for MI455X (gfx1250, wave32).
//   Stage A (scatter_kernel): agg[etype, dst, :] += x[src, :]  (fp32 global atomics)
//   Stage B (rgcn_fused):     per 16-node tile, all three GEMMs via
//                             V_WMMA_F32_16X16X4_F32, msg/mid staged in LDS.

typedef __attribute__((ext_vector_type(2))) float v2f;
typedef __attribute__((ext_vector_type(8))) float v8f;

#define DD 128
#define RR 8

// D = A(16x4,f32) * B(4x16,f32) + C(16x16,f32) on the wave32 matrix pipe.
__device__ __forceinline__ v8f wmma4(v2f a, v2f b, v8f c) {
  return __builtin_amdgcn_wmma_f32_16x16x4_f32(
      /*neg_a=*/false, a, /*neg_b=*/false, b,
      /*c_mod=*/(short)0, c, /*reuse_a=*/false, /*reuse_b=*/false);
}

// A-matrix 16x4 fp32 layout (ISA 7.12.2): lanes 0-15 hold M=lane, VGPR0=K0,VGPR1=K1;
// lanes 16-31 hold M=lane-16, VGPR0=K2,VGPR1=K3.
__device__ __forceinline__ v2f loadA(const float* __restrict__ A, int lda,
                                     int k0, int lane) {
  int r  = lane & 15;
  int kb = (lane >> 4) << 1;
  const float* p = A + (size_t)r * lda + (k0 + kb);
  v2f a;
  a.x = p[0];
  a.y = p[1];
  return a;
}

// B-matrix 4x16 fp32: row (K) striped across VGPRs, N striped across lanes:
// lanes 0-15 -> N=lane with K={k0,k0+1}; lanes 16-31 -> K={k0+2,k0+3}.
__device__ __forceinline__ v2f loadB(const float* __restrict__ B, int ldb,
                                     int k0, int n0, int lane) {
  int c  = lane & 15;
  int kb = (lane >> 4) << 1;
  const float* p = B + (size_t)(k0 + kb) * ldb + (n0 + c);
  v2f b;
  b.x = p[0];
  b.y = p[ldb];
  return b;
}

// -------- Stage A: edge scatter-add into agg[R, N, D] --------
// One wave per edge; each lane adds 4 floats (lane, lane+32, lane+64, lane+96).
__global__ __launch_bounds__(256) void scatter_kernel(
    const float* __restrict__ x, const int* __restrict__ src,
    const int* __restrict__ dst, const int* __restrict__ etype,
    float* __restrict__ agg, int E, int N) {
  int e = blockIdx.x * (blockDim.x >> 5) + (threadIdx.x >> 5);
  int lane = threadIdx.x & 31;
  if (e >= E) return;
  int s = src[e], d = dst[e], r = etype[e];
  const float* xs = x + (size_t)s * DD;
  float* ag = agg + ((size_t)r * N + d) * DD;
#pragma unroll
  for (int i = 0; i < 4; ++i) {
    int j = lane + 32 * i;
    atomicAdd(&ag[j], xs[j]);   // -> global_atomic_add_f32
  }
}

// -------- Stage B: fused per-node-tile GEMM chain --------
// Block = 256 threads = 8 waves; one block owns 16 node rows.
//  msg  (16x128): wave w computes col tile w*16 ; A from agg/x (global), B = W_rel/loop_w
//  mid  (16x256): wave w computes col tiles w*16 and (w+8)*16 ; A from x/s_msg
//  out  (16x128): wave w computes col tile w*16 ; A from x/s_mid
__global__ __launch_bounds__(256) void rgcn_fused(
    const float* __restrict__ x, const float* __restrict__ agg,
    const float* __restrict__ W_rel, const float* __restrict__ loop_w,
    const float* __restrict__ rel_bias, const float* __restrict__ W1,
    const float* __restrict__ b1, const float* __restrict__ W2,
    const float* __restrict__ b2, float* __restrict__ out, int N) {
  __shared__ float s_msg[16][DD + 1];       // +1 pad: kill LDS bank conflicts
  __shared__ float s_mid[16][2 * DD + 1];

  int lane = threadIdx.x & 31;
  int wave = threadIdx.x >> 5;
  int m0 = blockIdx.x * 16;
  const float* xa = x + (size_t)m0 * DD;

  // ---- msg tile: sum_r agg_r @ W_r + x @ loop_w + rel_bias ----
  {
    int n0 = wave * 16;
    v8f acc = {};
    for (int r = 0; r < RR; ++r) {
      const float* A = agg + ((size_t)r * N + m0) * DD;
      const float* B = W_rel + (size_t)r * DD * DD;
#pragma unroll 4
      for (int k = 0; k < DD; k += 4)
        acc = wmma4(loadA(A, DD, k, lane), loadB(B, DD, k, n0, lane), acc);
    }
#pragma unroll 4
    for (int k = 0; k < DD; k += 4)
      acc = wmma4(loadA(xa, DD, k, lane), loadB(loop_w, DD, k, n0, lane), acc);

    // C/D layout: lane L -> n = L&15, m = vgpr + (L>>4)*8
    int n = lane & 15, mb = (lane >> 4) << 3;
    float bias = rel_bias[n0 + n];
#pragma unroll
    for (int v = 0; v < 8; ++v) s_msg[mb + v][n0 + n] = acc[v] + bias;
  }
  __syncthreads();

  // ---- mid = tanh([x | msg] @ W1 + b1), 256 output cols ----
#pragma unroll
  for (int t = 0; t < 2; ++t) {
    int n0 = (wave + t * 8) * 16;
    v8f acc = {};
#pragma unroll 4
    for (int k = 0; k < DD; k += 4)  // x part (K = 0..127)
      acc = wmma4(loadA(xa, DD, k, lane), loadB(W1, 2 * DD, k, n0, lane), acc);
#pragma unroll 4
    for (int k = 0; k < DD; k += 4)  // msg part (K = 128..255), A from LDS
      acc = wmma4(loadA(&s_msg[0][0], DD + 1, k, lane),
                  loadB(W1, 2 * DD, DD + k, n0, lane), acc);
    int n = lane & 15, mb = (lane >> 4) << 3;
    float bias = b1[n0 + n];
#pragma unroll
    for (int v = 0; v < 8; ++v)
      s_mid[mb + v][n0 + n] = tanhf(acc[v] + bias);
  }
  __syncthreads();

  // ---- out = [x | mid] @ W2 + b2, 128 output cols, K = 384 ----
  {
    int n0 = wave * 16;
    v8f acc = {};
#pragma unroll 4
    for (int k = 0; k < DD; k += 4)  // x part
      acc = wmma4(loadA(xa, DD, k, lane), loadB(W2, DD, k, n0, lane), acc);
#pragma unroll 4
    for (int k = 0; k < 2 * DD; k += 4)  // mid part, A from LDS
      acc = wmma4(loadA(&s_mid[0][0], 2 * DD + 1, k, lane),
                  loadB(W2, DD, DD + k, n0, lane), acc);
    int n = lane & 15, mb = (lane >> 4) << 3;
    float bias = b2[n0 + n];
#pragma unroll
    for (int v = 0; v < 8; ++v)
      out[(size_t)(m0 + mb + v) * DD + (n0 + n)] = acc[v] + bias;
  }
}

extern "C" void kernel_launch(void* const* d_in, const int* in_sizes, int n_in,
                              void* d_out, int out_size, void* d_ws,
                              size_t ws_size, hipStream_t stream) {
  const float* x       = (const float*)d_in[0];
  const int*   src     = (const int*)d_in[1];
  const int*   dst     = (const int*)d_in[2];
  const int*   etype   = (const int*)d_in[3];
  const float* W_rel   = (const float*)d_in[4];
  const float* loop_w  = (const float*)d_in[5];
  const float* relbias = (const float*)d_in[6];
  const float* W1      = (const float*)d_in[7];
  const float* b1      = (const float*)d_in[8];
  const float* W2      = (const float*)d_in[9];
  const float* b2      = (const float*)d_in[10];
  float* out = (float*)d_out;

  int N = in_sizes[0] / DD;   // 100000
  int E = in_sizes[1];        // 1600000

  float* agg = (float*)d_ws;  // [R, N, D] fp32 = 409.6 MB
  size_t aggBytes = (size_t)RR * (size_t)N * DD * sizeof(float);
  hipMemsetAsync(agg, 0, aggBytes, stream);  // graph-capturable memset node

  // Stage A: 8 edges per 256-thread block (one wave per edge)
  int blocksA = (E + 7) / 8;
  scatter_kernel<<<blocksA, 256, 0, stream>>>(x, src, dst, etype, agg, E, N);

  // Stage B: one block per 16 node rows (N % 16 == 0 for this problem)
  int blocksB = N / 16;
  rgcn_fused<<<blocksB, 256, 0, stream>>>(x, agg, W_rel, loop_w, relbias, W1,
                                          b1, W2, b2, out, N);
}